// P2RLoss_83459804495938
// MI455X (gfx1250) — compile-verified
//
#include <hip/hip_runtime.h>
#include <math.h>

// Problem geometry (fixed by setup_inputs): B=4, H=W=128, N=2048, down=16.
// B and N are derived from in_sizes; H/W are compile-time.
#define HH 128
#define WW 128
#define HWSZ (HH * WW)            // 16384 pixels per image
#define TILE_M 16                 // pixels per WMMA tile (M dim)
#define WAVES_PER_BLOCK 8         // 256 threads / wave32
#define TILES_PER_IMG (HWSZ / TILE_M)              // 1024
#define BLOCKS_PER_IMG (TILES_PER_IMG / WAVES_PER_BLOCK) // 128

#define EPSV 1e-8f
#define R2 64.0f                  // MIN_RADIUS^2

typedef __attribute__((ext_vector_type(2))) float v2f;   // A/B operands: 16x4 / 4x16 f32 = 2 VGPRs
typedef __attribute__((ext_vector_type(8))) float v8f;   // C/D: 16x16 f32 = 8 VGPRs

// ---------------------------------------------------------------------------
// Kernel 1: per-image max(clip(den,0)) and sum(den). Deterministic LDS tree.
// ws[b] = max, ws[B+b] = sum
// ---------------------------------------------------------------------------
__global__ __launch_bounds__(256) void k_reduce_den(const float* __restrict__ dens,
                                                    float* __restrict__ ws, int B) {
  __shared__ float smax[256];
  __shared__ float ssum[256];
  const int b = blockIdx.x;
  const int tid = threadIdx.x;
  const float* dimg = dens + (size_t)b * HWSZ;
  float mx = 0.0f, sm = 0.0f;
  for (int i = tid; i < HWSZ; i += 256) {
    float v = dimg[i];
    sm += v;
    mx = fmaxf(mx, fmaxf(v, 0.0f));   // max of clipped map
  }
  smax[tid] = mx;
  ssum[tid] = sm;
  __syncthreads();
  for (int s = 128; s > 0; s >>= 1) {
    if (tid < s) {
      smax[tid] = fmaxf(smax[tid], smax[tid + s]);
      ssum[tid] += ssum[tid + s];
    }
    __syncthreads();
  }
  if (tid == 0) {
    ws[b] = smax[0];
    ws[B + b] = ssum[0];
  }
}

// ---------------------------------------------------------------------------
// Kernel 2: WMMA nearest-point distance + BCE per 16-pixel tile.
// One wave per pixel tile. d2[i][j] computed as a single 16x16x4 f32 WMMA:
//   A[i] = [c2_i, -2*rr_i, -2*cc_i, 1] ,  B[:,j] = [1, py_j, px_j, p2_j]
//
// The B operand is PRE-SWIZZLED into LDS in wave-register layout:
// LDS v2f element (j*32 + lane) is exactly what `lane` feeds the WMMA for
// point tile j:  lanes 0-15 -> {K0=1, K1=py},  lanes 16-31 -> {K2=px, K3=p2}.
// Inner loop = 1 ds_load_b64 + 1 v_wmma + 8 v_min per 16x16 tile of pairs.
//
// Per-tile BCE partial written to ws[2B + img*TILES_PER_IMG + tile].
// ---------------------------------------------------------------------------
__global__ __launch_bounds__(256) void k_p2r_tiles(const float* __restrict__ dens,
                                                   const float* __restrict__ points,
                                                   const int* __restrict__ downp,
                                                   float* __restrict__ ws,
                                                   int B, int N) {
  extern __shared__ float smem[];          // 4*N floats: N/16 tiles * 32 lanes * v2f
  v2f* s_b = (v2f*)smem;

  const int img  = blockIdx.x / BLOCKS_PER_IMG;
  const int blk  = blockIdx.x % BLOCKS_PER_IMG;
  const int tid  = threadIdx.x;
  const int lane = tid & 31;
  const int wave = tid >> 5;

  const float d = (float)(*downp);
  const float xclamp = (float)WW * d - 1.0f;
  const float yclamp = (float)HH * d - 1.0f;

  // Stage clamped points into LDS, directly in B-operand register layout.
  const float* pimg = points + (size_t)img * N * 2;
  for (int i = tid; i < N; i += blockDim.x) {
    float x = pimg[2 * i + 0];
    float y = pimg[2 * i + 1];
    x = fminf(fmaxf(x, 0.0f), xclamp);
    y = fminf(fmaxf(y, 0.0f), yclamp);
    const int j   = i >> 4;       // point tile
    const int col = i & 15;       // N column within tile
    v2f lo, hi;
    lo[0] = 1.0f;  lo[1] = y;             // lanes 0-15: K0, K1
    hi[0] = x;     hi[1] = x * x + y * y; // lanes 16-31: K2, K3
    s_b[j * 32 + col]      = lo;
    s_b[j * 32 + 16 + col] = hi;
  }
  __syncthreads();

  const int tile = blk * WAVES_PER_BLOCK + wave;      // [0, TILES_PER_IMG)
  const int row  = lane & 15;                          // M (lanes<16) / K-half select
  const int half = lane >> 4;                          // 0: K=0,1   1: K=2,3

  // A operand (loop invariant): 16x4 f32 layout —
  // lanes 0-15: V0=K0, V1=K1 ; lanes 16-31: V0=K2, V1=K3 (same M=row).
  const int pix = tile * TILE_M + row;
  const float rr = (float)(pix / WW) * d + (d - 1.0f) * 0.5f;
  const float cc = (float)(pix % WW) * d + (d - 1.0f) * 0.5f;
  v2f a;
  a[0] = half ? (-2.0f * cc) : (rr * rr + cc * cc);
  a[1] = half ? 1.0f         : (-2.0f * rr);

  v8f acc;
#pragma unroll
  for (int k = 0; k < 8; ++k) acc[k] = 3.0e38f;
  v8f czero = {};

  // Hot loop: ds_load_b64 -> wmma -> min-fold.
  const int ntiles = N / 16;
  const v2f* bptr = s_b + lane;
#pragma unroll 4
  for (int j = 0; j < ntiles; ++j) {
    v2f bm = bptr[j * 32];
    v8f dd = __builtin_amdgcn_wmma_f32_16x16x4_f32(
        /*neg_a=*/false, a, /*neg_b=*/false, bm,
        /*c_mod=*/(short)0, czero, /*reuse_a=*/false, /*reuse_b=*/false);
#pragma unroll
    for (int k = 0; k < 8; ++k) acc[k] = fminf(acc[k], dd[k]);
  }

  // Min over N: reduce across each 16-lane half (C/D layout: lane group = N).
  float mins[8];
#pragma unroll
  for (int k = 0; k < 8; ++k) {
    float v = acc[k];
    v = fminf(v, __shfl_xor(v, 1));
    v = fminf(v, __shfl_xor(v, 2));
    v = fminf(v, __shfl_xor(v, 4));
    v = fminf(v, __shfl_xor(v, 8));
    mins[k] = v;   // lanes 0-15: pixels tile*16+0..7 ; lanes 16-31: +8..15
  }

  // BCE for this tile's 16 pixels: lane 0 handles M=0..7, lane 16 handles M=8..15.
  const float mx = ws[img];
  float lsum = 0.0f;
  if (row == 0) {
    const float* dimg = dens + (size_t)img * HWSZ;
    const int pbase = tile * TILE_M + half * 8;
#pragma unroll
    for (int m = 0; m < 8; ++m) {
      const float denv = dimg[pbase + m];
      const float x = (mx > 0.0f) ? (fmaxf(denv, 0.0f) / (mx + EPSV)) : 0.0f;
      // minC < 8  <=>  clip(d2,0) < 64  <=>  d2 < 64
      const float t = (mins[m] < R2) ? 1.0f : 0.0f;
      const float w = t + 1.0f;
      const float logx   = (x > 0.0f) ? logf(x)      : -100.0f;
      const float log1mx = (x < 1.0f) ? log1pf(-x)   : -100.0f;
      lsum += -w * (t * logx + (1.0f - t) * log1mx);
    }
  }
  lsum += __shfl_xor(lsum, 16);               // lane0 += lane16
  if (lane == 0) ws[2 * B + img * TILES_PER_IMG + tile] = lsum;
}

// ---------------------------------------------------------------------------
// Kernel 3: deterministic fixed-order final reduction -> scalar loss.
// ---------------------------------------------------------------------------
__global__ void k_finalize(const float* __restrict__ ws, float* __restrict__ out,
                           const int* __restrict__ downp, int B, int N) {
  if (threadIdx.x == 0 && blockIdx.x == 0) {
    const float d = (float)(*downp);
    float per_sum = 0.0f, pen = 0.0f;
    for (int b = 0; b < B; ++b) {
      float s = 0.0f;
      const float* part = ws + 2 * B + (size_t)b * TILES_PER_IMG;
      for (int t = 0; t < TILES_PER_IMG; ++t) s += part[t];
      per_sum += s / (float)HWSZ;                    // mean BCE per image
      const float pred = ws[B + b] / (d * d);        // sum(den)/down^2
      pen += fabsf(pred - (float)N);
    }
    pen = 0.02f * (pen / (float)B);                  // SCALE_WEIGHT * mean|.|
    out[0] = (per_sum + pen) / (float)B;
  }
}

// ---------------------------------------------------------------------------
extern "C" void kernel_launch(void* const* d_in, const int* in_sizes, int n_in,
                              void* d_out, int out_size, void* d_ws, size_t ws_size,
                              hipStream_t stream) {
  const float* dens   = (const float*)d_in[0];   // [B,1,H,W] f32
  const float* points = (const float*)d_in[1];   // [B,N,2]   f32
  const int*   downp  = (const int*)d_in[2];     // scalar int
  float* out = (float*)d_out;
  float* ws  = (float*)d_ws;

  const int B = in_sizes[0] / HWSZ;              // 4
  const int N = in_sizes[1] / (2 * B);           // 2048

  k_reduce_den<<<B, 256, 0, stream>>>(dens, ws, B);

  const size_t shmem = (size_t)4 * N * sizeof(float);   // 32 KB: pre-swizzled B operands
  k_p2r_tiles<<<B * BLOCKS_PER_IMG, 256, shmem, stream>>>(dens, points, downp, ws, B, N);

  k_finalize<<<1, 32, 0, stream>>>(ws, out, downp, B, N);
}